// GNNClassifier_24790551232826
// MI455X (gfx1250) — compile-verified
//
#include <hip/hip_runtime.h>
#include <math.h>

// ---------------- problem constants (match reference) ----------------
#define NN   50000      // nodes (divisible by 16 -> 3125 row tiles)
#define NE   800000     // edges
#define DIM  64         // IN_DIM == HID_DIM
#define NCLS 10
#define NCP  16         // padded classes
#define NTILES (NN / 16)   // 3125

typedef __attribute__((ext_vector_type(2))) float v2f;
typedef __attribute__((ext_vector_type(8))) float v8f;

__device__ __forceinline__ void atomAddF(float* p, float v) {
  __hip_atomic_fetch_add(p, v, __ATOMIC_RELAXED, __HIP_MEMORY_SCOPE_AGENT);
}

// ---------------- degree / norm ----------------
__global__ void k_deg_init(float* __restrict__ deg) {
  int i = blockIdx.x * blockDim.x + threadIdx.x;
  if (i < NN) deg[i] = 1.0f;                    // self loop contributes 1
}

__global__ void k_deg_edges(const long long* __restrict__ dst, float* __restrict__ deg) {
  int e = blockIdx.x * blockDim.x + threadIdx.x;
  if (e < NE) atomAddF(&deg[(int)dst[e]], 1.0f);
}

__global__ void k_dinv(float* __restrict__ deg) {
  int i = blockIdx.x * blockDim.x + threadIdx.x;
  if (i < NN) deg[i] = rsqrtf(deg[i]);          // deg >= 1 always
}

// ---------------- GEMM1: xl = x @ W1  (fp32 WMMA 16x16x4) ----------------
// One wave computes a 16-row tile across all 64 output columns.
__global__ __launch_bounds__(256) void k_gemm1(const float* __restrict__ x,
                                               const float* __restrict__ W,
                                               float* __restrict__ xl) {
  __shared__ float lW[DIM * DIM];               // 16 KB
  int t = threadIdx.x;
  for (int i = t * 4; i < DIM * DIM; i += 256 * 4)
    *(float4*)(&lW[i]) = *(const float4*)(&W[i]);
  __syncthreads();

  int wave = t >> 5;
  int lane = t & 31;
  int rowTile = blockIdx.x * 8 + wave;
  if (rowTile >= NTILES) return;                // whole-wave uniform: EXEC stays all-1s

  int m0    = rowTile * 16;
  int lrow  = lane & 15;
  int khalf = lane >> 4;                        // ISA layout: lanes 16-31 hold K+2
  int row   = m0 + lrow;

  v8f acc[4] = {};
  for (int k = 0; k < DIM; k += 4) {
    int kb = k + 2 * khalf;
    v2f a = *(const v2f*)(&x[(size_t)row * DIM + kb]);   // A: 16x4, K pair per lane
#pragma unroll
    for (int j = 0; j < 4; ++j) {
      int col = j * 16 + lrow;
      v2f b;
      b.x = lW[kb * DIM + col];                 // B: 4x16, K pair per lane
      b.y = lW[(kb + 1) * DIM + col];
      acc[j] = __builtin_amdgcn_wmma_f32_16x16x4_f32(
          false, a, false, b, (short)0, acc[j], false, false);
    }
  }
  // C/D layout: VGPR v -> row m0 + v + 8*khalf, col = lrow (+ tile offset)
#pragma unroll
  for (int j = 0; j < 4; ++j) {
    int col = j * 16 + lrow;
#pragma unroll
    for (int v = 0; v < 8; ++v)
      xl[(size_t)(m0 + v + 8 * khalf) * DIM + col] = acc[j][v];
  }
}

// ---------------- layer-1 aggregation ----------------
__global__ void k_h_init(const float* __restrict__ xl, const float* __restrict__ dinv,
                         const float* __restrict__ b1, float* __restrict__ h) {
  long long i = (long long)blockIdx.x * blockDim.x + threadIdx.x;
  if (i >= (long long)NN * DIM) return;
  int node = (int)(i >> 6), c = (int)(i & 63);
  float sw = dinv[node] * dinv[node];           // self-loop norm = 1/deg
  h[i] = b1[c] + sw * xl[i];
}

__global__ void k_agg1(const long long* __restrict__ src, const long long* __restrict__ dst,
                       const float* __restrict__ dinv, const float* __restrict__ xl,
                       float* __restrict__ h) {
  long long gid = (long long)blockIdx.x * blockDim.x + threadIdx.x;
  long long e = gid >> 4;                       // 16 threads per edge, 4 cols each
  if (e >= NE) return;
  int q = (int)(gid & 15) * 4;
  int s = (int)src[e], d = (int)dst[e];
  float w = dinv[s] * dinv[d];
  float4 v = *(const float4*)(&xl[(size_t)s * DIM + q]);
  float* hp = &h[(size_t)d * DIM + q];
  atomAddF(hp + 0, w * v.x);
  atomAddF(hp + 1, w * v.y);
  atomAddF(hp + 2, w * v.z);
  atomAddF(hp + 3, w * v.w);
}

// ---------------- GEMM2: xl2 = relu(h) @ W2pad  (fp32 WMMA) ----------------
__global__ __launch_bounds__(256) void k_gemm2(const float* __restrict__ h,
                                               const float* __restrict__ W2,
                                               float* __restrict__ xl2) {
  __shared__ float lW[DIM * NCP];               // 64x16, zero-padded cols 10..15
  int t = threadIdx.x;
  for (int i = t; i < DIM * NCP; i += 256) {
    int kk = i >> 4, cc = i & 15;
    lW[i] = (cc < NCLS) ? W2[kk * NCLS + cc] : 0.0f;
  }
  __syncthreads();

  int wave = t >> 5;
  int lane = t & 31;
  int rowTile = blockIdx.x * 8 + wave;
  if (rowTile >= NTILES) return;

  int m0    = rowTile * 16;
  int lrow  = lane & 15;
  int khalf = lane >> 4;
  int row   = m0 + lrow;

  v8f acc = {};
  for (int k = 0; k < DIM; k += 4) {
    int kb = k + 2 * khalf;
    v2f a;                                      // ReLU folded into the A load
    a.x = fmaxf(h[(size_t)row * DIM + kb], 0.0f);
    a.y = fmaxf(h[(size_t)row * DIM + kb + 1], 0.0f);
    v2f b;
    b.x = lW[kb * NCP + lrow];
    b.y = lW[(kb + 1) * NCP + lrow];
    acc = __builtin_amdgcn_wmma_f32_16x16x4_f32(
        false, a, false, b, (short)0, acc, false, false);
  }
#pragma unroll
  for (int v = 0; v < 8; ++v)
    xl2[(size_t)(m0 + v + 8 * khalf) * NCP + lrow] = acc[v];
}

// ---------------- layer-2 aggregation ----------------
__global__ void k_logits_init(const float* __restrict__ xl2, const float* __restrict__ dinv,
                              const float* __restrict__ b2, float* __restrict__ logits) {
  long long i = (long long)blockIdx.x * blockDim.x + threadIdx.x;
  if (i >= (long long)NN * NCP) return;
  int node = (int)(i >> 4), c = (int)(i & 15);
  float sw = dinv[node] * dinv[node];
  float bias = (c < NCLS) ? b2[c] : 0.0f;
  logits[i] = bias + sw * xl2[i];
}

__global__ void k_agg2(const long long* __restrict__ src, const long long* __restrict__ dst,
                       const float* __restrict__ dinv, const float* __restrict__ xl2,
                       float* __restrict__ logits) {
  long long gid = (long long)blockIdx.x * blockDim.x + threadIdx.x;
  long long e = gid >> 2;                       // 4 threads per edge, 4 cols each
  if (e >= NE) return;
  int q = (int)(gid & 3) * 4;
  int s = (int)src[e], d = (int)dst[e];
  float w = dinv[s] * dinv[d];
  float4 v = *(const float4*)(&xl2[(size_t)s * NCP + q]);
  float* lp = &logits[(size_t)d * NCP + q];
  atomAddF(lp + 0, w * v.x);
  atomAddF(lp + 1, w * v.y);
  atomAddF(lp + 2, w * v.z);
  atomAddF(lp + 3, w * v.w);
}

// ---------------- log_softmax over 10 classes ----------------
__global__ void k_logsoftmax(const float* __restrict__ logits, float* __restrict__ out) {
  int node = blockIdx.x * blockDim.x + threadIdx.x;
  if (node >= NN) return;
  const float* lp = &logits[(size_t)node * NCP];
  float m = lp[0];
#pragma unroll
  for (int c = 1; c < NCLS; ++c) m = fmaxf(m, lp[c]);
  float ssum = 0.0f;
#pragma unroll
  for (int c = 0; c < NCLS; ++c) ssum += expf(lp[c] - m);
  float lse = logf(ssum);
  float* op = &out[(size_t)node * NCLS];
#pragma unroll
  for (int c = 0; c < NCLS; ++c) op[c] = lp[c] - m - lse;
}

// ---------------- launch ----------------
extern "C" void kernel_launch(void* const* d_in, const int* in_sizes, int n_in,
                              void* d_out, int out_size, void* d_ws, size_t ws_size,
                              hipStream_t stream) {
  const float*     x   = (const float*)d_in[0];            // [NN, 64]
  const long long* ei  = (const long long*)d_in[1];        // [2, NE] int64
  const float*     W1  = (const float*)d_in[2];            // [64, 64]
  const float*     b1  = (const float*)d_in[3];            // [64]
  const float*     W2  = (const float*)d_in[4];            // [64, 10]
  const float*     b2  = (const float*)d_in[5];            // [10]
  float*           out = (float*)d_out;                    // [NN, 10]

  const long long* src = ei;                               // row 0
  const long long* dst = ei + NE;                          // row 1

  // workspace carve: dinv(N) | xl(N*64) | h(N*64) | xl2(N*16) | logits(N*16)
  float* ws     = (float*)d_ws;
  float* dinv   = ws;
  float* xl     = dinv + NN;
  float* h      = xl + (size_t)NN * DIM;
  float* xl2    = h + (size_t)NN * DIM;
  float* logits = xl2 + (size_t)NN * NCP;

  const int B = 256;
  // 1) degrees -> dinv (in place)
  k_deg_init<<<(NN + B - 1) / B, B, 0, stream>>>(dinv);
  k_deg_edges<<<(NE + B - 1) / B, B, 0, stream>>>(dst, dinv);
  k_dinv<<<(NN + B - 1) / B, B, 0, stream>>>(dinv);

  // 2) xl = x @ W1 (WMMA fp32)
  k_gemm1<<<(NTILES + 7) / 8, 256, 0, stream>>>(x, W1, xl);

  // 3) h = b1 + selfloop + scatter-add over edges
  k_h_init<<<(int)(((long long)NN * DIM + B - 1) / B), B, 0, stream>>>(xl, dinv, b1, h);
  k_agg1<<<(int)(((long long)NE * 16 + B - 1) / B), B, 0, stream>>>(src, dst, dinv, xl, h);

  // 4) xl2 = relu(h) @ W2 (WMMA fp32, classes padded to 16)
  k_gemm2<<<(NTILES + 7) / 8, 256, 0, stream>>>(h, W2, xl2);

  // 5) logits = b2 + selfloop + scatter-add
  k_logits_init<<<(int)(((long long)NN * NCP + B - 1) / B), B, 0, stream>>>(xl2, dinv, b2, logits);
  k_agg2<<<(int)(((long long)NE * 4 + B - 1) / B), B, 0, stream>>>(src, dst, dinv, xl2, logits);

  // 6) log_softmax -> d_out
  k_logsoftmax<<<(NN + B - 1) / B, B, 0, stream>>>(logits, out);
}